// fs_net_loss_12463995093445
// MI455X (gfx1250) — compile-verified
//
#include <hip/hip_runtime.h>
#include <math.h>

// ---- types ----------------------------------------------------------------
typedef __attribute__((ext_vector_type(2))) float v2f;
typedef __attribute__((ext_vector_type(4))) float f4;
typedef __attribute__((ext_vector_type(8))) float v8f;

#define BS_      128
#define NPTS_    32768
#define BASE_    12
#define LOSS_W   8.0f
#define RBLOCKS  1024
#define RTHREADS 256

// ---- wave32 reduction through the matrix pipe -----------------------------
// 32 lane partials -> total (broadcast to all lanes of the wave).
// A (16x4 f32): VGPR0 holds lane partials (lanes0-15 -> A[m][0], lanes16-31 ->
// A[m][2]); VGPR1 = 0 (K=1,3). B = ones(4x16). D[m][n] = p[m] + p[m+16].
// D layout: VGPR r, lanes0-15 = row r, lanes16-31 = row r+8. Summing the 8 D
// VGPRs per lane leaves (sum p[0..7]+p[16..23]) on lanes<16 and
// (sum p[8..15]+p[24..31]) on lanes>=16; one shfl_xor(16) finishes.
__device__ __forceinline__ float wave_reduce_wmma(float v) {
    v2f a; a.x = v;    a.y = 0.0f;
    v2f b; b.x = 1.0f; b.y = 1.0f;
    v8f c = {};
    v8f d = __builtin_amdgcn_wmma_f32_16x16x4_f32(
        /*neg_a=*/false, a, /*neg_b=*/false, b,
        /*c_mod=*/(short)0, c, /*reuse_a=*/false, /*reuse_b=*/false);
    float t = d[0] + d[1] + d[2] + d[3] + d[4] + d[5] + d[6] + d[7];
    t += __shfl_xor(t, 16, 32);
    return t;
}

// 256-thread block reduction; result valid on threads 0..31 (wave 0).
__device__ __forceinline__ float block_reduce_256(float v, float* sm) {
    const int tid = threadIdx.x;
    sm[tid] = v;                      __syncthreads();
    if (tid < 128) sm[tid] += sm[tid + 128]; __syncthreads();
    if (tid <  64) sm[tid] += sm[tid +  64]; __syncthreads();
    if (tid <  32) sm[tid] += sm[tid +  32]; __syncthreads();
    float r = 0.0f;
    if (tid < 32) r = wave_reduce_wmma(sm[tid]);   // uniform branch per wave
    __syncthreads();
    return r;
}

// ---- kernel 1: streaming |pred_Recon - gt_Recon| partial sums -------------
__global__ __launch_bounds__(RTHREADS)
void recon_reduce_kernel(const f4* __restrict__ p, const f4* __restrict__ g,
                         float* __restrict__ partials, int n4) {
    __shared__ float sm[RTHREADS];
    const int tid    = threadIdx.x;
    const int stride = gridDim.x * blockDim.x;
    float acc = 0.0f;
    for (int i = blockIdx.x * blockDim.x + tid; i < n4; i += stride) {
        f4 a = __builtin_nontemporal_load(&p[i]);   // global_load_b128, NT
        f4 b = __builtin_nontemporal_load(&g[i]);
        acc += fabsf(a.x - b.x) + fabsf(a.y - b.y)
             + fabsf(a.z - b.z) + fabsf(a.w - b.w);
    }
    float tot = block_reduce_256(acc, sm);
    if (tid == 0) partials[blockIdx.x] = tot;
}

// ---- kernel 2: small losses + finalize ------------------------------------
__global__ __launch_bounds__(256)
void fsnet_finalize_kernel(const float* __restrict__ pR1,
                           const float* __restrict__ pR2,
                           const float* __restrict__ pT,
                           const float* __restrict__ pS,
                           const float* __restrict__ gR,
                           const float* __restrict__ gT,
                           const float* __restrict__ gS,
                           const int*   __restrict__ sym,
                           const float* __restrict__ partials, int n_partials,
                           float* __restrict__ out) {
    __shared__ float sm[256];
    const int tid = threadIdx.x;

    // recon: fold 1024 block partials into 256 lanes
    float rsum = 0.0f;
    for (int i = tid; i < n_partials; i += 256) rsum += partials[i];

    float rot1_b = 0.0f, l2m_b = 0.0f, mask_b = 0.0f, tran_b = 0.0f, size_b = 0.0f;
    if (tid < BS_) {
        const int b = tid;
        float R[9];
        #pragma unroll
        for (int i = 0; i < 9; ++i) R[i] = gR[b * 9 + i];

        // rot1: canonical = R @ [0,1,0] = column 1 of R (and its negation)
        const float c0 = R[1], c1 = R[4], c2 = R[7];
        const float p0 = pR1[b*3+0], p1 = pR1[b*3+1], p2 = pR1[b*3+2];
        const float l1a = (fabsf(p0 - c0) + fabsf(p1 - c1) + fabsf(p2 - c2)) * (1.0f/3.0f);
        const float l1b = (fabsf(p0 + c0) + fabsf(p1 + c1) + fabsf(p2 + c2)) * (1.0f/3.0f);
        rot1_b = (sym[b*6+2] == 1) ? fminf(l1a, l1b) : l1a;

        // rot2: min over BASE y-rotations; R @ (Ry(th) @ [1,0,0]) = c*col0 - s*col2
        const float q0 = pR2[b*3+0], q1 = pR2[b*3+1], q2 = pR2[b*3+2];
        float best = 3.4e38f;
        #pragma unroll
        for (int k = 0; k < BASE_; ++k) {
            const float th = ((float)k / (float)BASE_) * 6.28318530717958647692f;
            const float c = cosf(th), s = sinf(th);
            const float a0 = c * R[0] - s * R[2];
            const float a1 = c * R[3] - s * R[5];
            const float a2 = c * R[6] - s * R[8];
            const float l = (fabsf(q0 - a0) + fabsf(q1 - a1) + fabsf(q2 - a2)) * (1.0f/3.0f);
            best = fminf(best, l);
        }
        mask_b = (sym[b*6+0] == 0) ? 1.0f : 0.0f;
        l2m_b  = best * mask_b;

        tran_b = fabsf(pT[b*3+0] - gT[b*3+0]) + fabsf(pT[b*3+1] - gT[b*3+1])
               + fabsf(pT[b*3+2] - gT[b*3+2]);
        size_b = fabsf(pS[b*3+0] - gS[b*3+0]) + fabsf(pS[b*3+1] - gS[b*3+1])
               + fabsf(pS[b*3+2] - gS[b*3+2]);
    }

    const float tot_recon = block_reduce_256(rsum,   sm);
    const float tot_rot1  = block_reduce_256(rot1_b, sm);
    const float tot_l2m   = block_reduce_256(l2m_b,  sm);
    const float tot_mask  = block_reduce_256(mask_b, sm);
    const float tot_tran  = block_reduce_256(tran_b, sm);
    const float tot_size  = block_reduce_256(size_b, sm);

    if (tid == 0) {
        out[0] = LOSS_W * (tot_rot1 * (1.0f / (float)BS_));
        const float rot2 = (tot_mask > 0.0f)
                         ? (tot_l2m / fmaxf(tot_mask, 1.0f)) : 0.0f;
        out[1] = LOSS_W * rot2;
        out[2] = LOSS_W * (tot_recon * (1.0f / (float)(BS_ * NPTS_ * 3)));
        out[3] = LOSS_W * (tot_tran  * (1.0f / (float)(BS_ * 3)));
        out[4] = LOSS_W * (tot_size  * (1.0f / (float)(BS_ * 3)));
    }
}

// ---- launcher -------------------------------------------------------------
extern "C" void kernel_launch(void* const* d_in, const int* in_sizes, int n_in,
                              void* d_out, int out_size, void* d_ws, size_t ws_size,
                              hipStream_t stream) {
    const float* pR1    = (const float*)d_in[0];
    const float* pR2    = (const float*)d_in[1];
    const float* pRecon = (const float*)d_in[2];
    const float* pT     = (const float*)d_in[3];
    const float* pS     = (const float*)d_in[4];
    const float* gR     = (const float*)d_in[5];
    const float* gRecon = (const float*)d_in[6];
    const float* gT     = (const float*)d_in[7];
    const float* gS     = (const float*)d_in[8];
    const int*   sym    = (const int*)  d_in[9];

    float* partials = (float*)d_ws;      // RBLOCKS floats, fully rewritten per call
    float* out      = (float*)d_out;     // 5 floats

    const int n4 = in_sizes[2] / 4;      // 12,582,912 / 4 f4-pairs

    recon_reduce_kernel<<<RBLOCKS, RTHREADS, 0, stream>>>(
        (const f4*)pRecon, (const f4*)gRecon, partials, n4);

    fsnet_finalize_kernel<<<1, 256, 0, stream>>>(
        pR1, pR2, pT, pS, gR, gT, gS, sym, partials, RBLOCKS, out);
}